// CRF_8976481648991
// MI455X (gfx1250) — compile-verified
//
#include <hip/hip_runtime.h>
#include <math.h>

// CRF forward algorithm as a per-timestep GEMM on CDNA5 WMMA.
// Sizes fixed by the reference problem.
#define CRF_B 256
#define CRF_T 1024
#define CRF_K 128
#define ROWS  16      // batch rows per workgroup (WMMA M)
#define PITCH 132     // 128 + 4 floats pad -> conflict-free LDS row access

typedef float v2f __attribute__((ext_vector_type(2)));
typedef float v8f __attribute__((ext_vector_type(8)));

// ---------------------------------------------------------------------------
// Kernel 1: forward recursion. One WG = 16 batch rows, 8 waves = 8 N-tiles.
// Per step: la_next[j] = e_t[j] + m + log( sum_i exp(trans[i,j]) * exp(la[i]-m) )
// The inner sum is a 16x128 @ 128x16 f32 WMMA chain (K-loop of 32 x 16x16x4).
// ---------------------------------------------------------------------------
__global__ __launch_bounds__(256) void crf_forward_kernel(
    const float* __restrict__ emissions, const int* __restrict__ mask,
    const float* __restrict__ transitions, const float* __restrict__ start_t,
    const float* __restrict__ end_t, float* __restrict__ logz)
{
  __shared__ __align__(16) float la[ROWS * PITCH];   // log-alpha
  __shared__ __align__(16) float ae[ROWS * PITCH];   // exp(la - rowmax)
  __shared__ float mrow[ROWS];

  const int tid   = threadIdx.x;
  const int wave  = tid >> 5;
  const int lane  = tid & 31;
  const int lhalf = lane >> 4;   // 0 | 1
  const int l15   = lane & 15;
  const int b0    = blockIdx.x * ROWS;
  const int n0    = wave * 16;   // this wave's output-column tile

  // B fragments resident in VGPRs for the entire T loop:
  // E[i][j] = exp(transitions[i][j]), columns n0..n0+15.
  // V_WMMA_F32_16X16X4_F32 B (4x16): .x -> K = 4s + 2*lhalf, .y -> K+1, N = n0+l15.
  v2f Bf[32];
#pragma unroll
  for (int s = 0; s < 32; ++s) {
    const int r0 = 4 * s + 2 * lhalf;
    Bf[s].x = __expf(transitions[r0 * CRF_K + n0 + l15]);
    Bf[s].y = __expf(transitions[(r0 + 1) * CRF_K + n0 + l15]);
  }

  // t = 0 init: log_alpha = start_transitions + emissions[:, 0, :]
  for (int idx = tid; idx < ROWS * CRF_K; idx += 256) {
    const int r = idx >> 7, c = idx & 127;
    la[r * PITCH + c] = start_t[c] + emissions[((size_t)(b0 + r)) * CRF_T * CRF_K + c];
  }
  __syncthreads();

  for (int t = 1; t < CRF_T; ++t) {
    // (1) per-row max over K (16 threads/row, wave32 xor butterfly)
    {
      const int r = tid >> 4, c = tid & 15;
      float mx = -3.4e38f;
      for (int c2 = c; c2 < CRF_K; c2 += 16) mx = fmaxf(mx, la[r * PITCH + c2]);
#pragma unroll
      for (int off = 8; off >= 1; off >>= 1) mx = fmaxf(mx, __shfl_xor(mx, off, 32));
      if (c == 0) mrow[r] = mx;
    }
    __syncthreads();

    // (2) a = exp(la - m)
    for (int idx = tid; idx < ROWS * CRF_K; idx += 256) {
      const int r = idx >> 7, c = idx & 127;
      ae[r * PITCH + c] = __expf(la[r * PITCH + c] - mrow[r]);
    }
    __syncthreads();

    // (3) acc[r][n0+j] = sum_i a[r][i] * E[i][n0+j]  via 32x V_WMMA_F32_16X16X4_F32
    v8f acc = {};
#pragma unroll
    for (int s = 0; s < 32; ++s) {
      // A (16x4): lane l -> M = l&15, K = 4s + 2*lhalf + {0,1}; 8B-aligned LDS read.
      const v2f a = *(const v2f*)&ae[l15 * PITCH + 4 * s + 2 * lhalf];
      acc = __builtin_amdgcn_wmma_f32_16x16x4_f32(
          /*neg_a=*/false, a, /*neg_b=*/false, Bf[s],
          /*c_mod=*/(short)0, acc, /*reuse_a=*/false, /*reuse_b=*/false);
    }

    // (4) la_next = mask ? (e_t + m + log(acc)) : la
    // C/D layout: VGPR p -> row M = p + 8*lhalf, col N = n0 + l15.
#pragma unroll
    for (int p = 0; p < 8; ++p) {
      const int r = p + 8 * lhalf;
      const int b = b0 + r;
      const int j = n0 + l15;
      const float* ep = &emissions[((size_t)b * CRF_T + t) * CRF_K + j];
      __builtin_prefetch(ep + CRF_K, 0, 1);   // stream next timestep's emission
      const float lnew = *ep + mrow[r] + __logf(acc[p]);
      const float lold = la[r * PITCH + j];
      const float mv   = (float)mask[b * CRF_T + t];
      la[r * PITCH + j] = mv * lnew + (1.f - mv) * lold;
    }
    __syncthreads();
  }

  // logZ[b] = logsumexp_k( la[b][k] + end_transitions[k] )
  {
    const int r = tid >> 4, c = tid & 15;
    float mx = -3.4e38f;
    for (int c2 = c; c2 < CRF_K; c2 += 16) mx = fmaxf(mx, la[r * PITCH + c2] + end_t[c2]);
#pragma unroll
    for (int off = 8; off >= 1; off >>= 1) mx = fmaxf(mx, __shfl_xor(mx, off, 32));
    float sum = 0.f;
    for (int c2 = c; c2 < CRF_K; c2 += 16) sum += __expf(la[r * PITCH + c2] + end_t[c2] - mx);
#pragma unroll
    for (int off = 8; off >= 1; off >>= 1) sum += __shfl_xor(sum, off, 32);
    if (c == 0) logz[b0 + r] = mx + __logf(sum);
  }
}

// ---------------------------------------------------------------------------
// Kernel 2: gold-path score per batch. One block per sequence.
// ---------------------------------------------------------------------------
__global__ __launch_bounds__(256) void crf_score_kernel(
    const float* __restrict__ emissions, const int* __restrict__ tags,
    const int* __restrict__ mask, const float* __restrict__ transitions,
    const float* __restrict__ start_t, const float* __restrict__ end_t,
    float* __restrict__ score)
{
  __shared__ float sbuf[256];
  __shared__ int   ibuf[256];
  const int b = blockIdx.x;
  const int tid = threadIdx.x;
  float acc = 0.f;
  int msum = 0;
  for (int t = tid; t < CRF_T; t += 256) {
    const int tg = tags[b * CRF_T + t];
    const int mk = mask[b * CRF_T + t];
    acc += emissions[((size_t)b * CRF_T + t) * CRF_K + tg] * (float)mk;
    if (t >= 1) {
      const int tgp = tags[b * CRF_T + t - 1];
      acc += transitions[tgp * CRF_K + tg] * (float)mk;
    }
    msum += mk;
  }
  sbuf[tid] = acc; ibuf[tid] = msum;
  __syncthreads();
  for (int s2 = 128; s2 > 0; s2 >>= 1) {
    if (tid < s2) { sbuf[tid] += sbuf[tid + s2]; ibuf[tid] += ibuf[tid + s2]; }
    __syncthreads();
  }
  if (tid == 0) {
    const int last = ibuf[0] - 1;  // seq_end index
    score[b] = sbuf[0] + start_t[tags[b * CRF_T]] + end_t[tags[b * CRF_T + last]];
  }
}

// ---------------------------------------------------------------------------
// Kernel 3: out = mean_b( logZ[b] - score[b] )  ( = -(score - logZ).mean() )
// ---------------------------------------------------------------------------
__global__ __launch_bounds__(256) void crf_final_kernel(
    const float* __restrict__ logz, const float* __restrict__ score,
    float* __restrict__ out)
{
  __shared__ float buf[256];
  const int tid = threadIdx.x;
  buf[tid] = logz[tid] - score[tid];
  __syncthreads();
  for (int s2 = 128; s2 > 0; s2 >>= 1) {
    if (tid < s2) buf[tid] += buf[tid + s2];
    __syncthreads();
  }
  if (tid == 0) out[0] = buf[0] / (float)CRF_B;
}

extern "C" void kernel_launch(void* const* d_in, const int* in_sizes, int n_in,
                              void* d_out, int out_size, void* d_ws, size_t ws_size,
                              hipStream_t stream) {
  const float* emissions   = (const float*)d_in[0];   // (B,T,K) f32
  const int*   tags        = (const int*)  d_in[1];   // (B,T) i32
  const int*   mask        = (const int*)  d_in[2];   // (B,T) i32
  const float* transitions = (const float*)d_in[3];   // (K,K) f32
  const float* start_t     = (const float*)d_in[4];   // (K,) f32
  const float* end_t       = (const float*)d_in[5];   // (K,) f32
  float* out = (float*)d_out;

  float* logz  = (float*)d_ws;        // 256 floats
  float* score = logz + CRF_B;        // 256 floats

  crf_forward_kernel<<<CRF_B / ROWS, 256, 0, stream>>>(emissions, mask, transitions,
                                                       start_t, end_t, logz);
  crf_score_kernel<<<CRF_B, 256, 0, stream>>>(emissions, tags, mask, transitions,
                                              start_t, end_t, score);
  crf_final_kernel<<<1, 256, 0, stream>>>(logz, score, out);
}